// TopKActivation_81776177316427
// MI455X (gfx1250) — compile-verified
//
#include <hip/hip_runtime.h>

#ifndef __has_builtin
#define __has_builtin(x) 0
#endif

#define TPB    1024
#define COLS   32768
#define CHUNKS ((COLS / 4) / TPB)   // 8 x 16B per thread
#define EPT    (COLS / TPB)         // 32 elements per thread
#define NHIST  16                   // privatized LDS histograms (2 waves each)

typedef int   v4i __attribute__((vector_size(16)));
typedef float v4f __attribute__((vector_size(16)));
typedef __attribute__((address_space(1))) v4i* g_v4i_ptr;
typedef __attribute__((address_space(3))) v4i* l_v4i_ptr;

__global__ __launch_bounds__(TPB) void topk64_relu_kernel(
    const float* __restrict__ x, const int* __restrict__ kptr,
    float* __restrict__ out) {
  extern __shared__ unsigned char smem[];
  float*    row   = reinterpret_cast<float*>(smem);                       // 128 KB
  unsigned* whist = reinterpret_cast<unsigned*>(smem + COLS * 4);         // 16 KB
  unsigned* scal  = reinterpret_cast<unsigned*>(smem + COLS * 4 + NHIST * 256 * 4);
  // scal[0]=prefix/threshold bits  scal[1]=need/remaining  scal[2]=done  scal[3]=tie counter

  const int    tid     = threadIdx.x;
  const size_t rowBase = (size_t)blockIdx.x * COLS;
  const float* gx      = x + rowBase;
  float*       gout    = out + rowBase;

  // ---------------- stage one row (128 KB) into LDS ----------------
#if __has_builtin(__builtin_amdgcn_global_load_async_to_lds_b128)
  #pragma unroll
  for (int c = 0; c < CHUNKS; ++c) {
    const int f4 = c * TPB + tid;
    __builtin_amdgcn_global_load_async_to_lds_b128(
        (g_v4i_ptr)(gx + 4 * f4), (l_v4i_ptr)(row + 4 * f4), 0, 0);
  }
  #if __has_builtin(__builtin_amdgcn_s_wait_asynccnt)
  __builtin_amdgcn_s_wait_asynccnt(0);
  #else
  asm volatile("s_wait_asynccnt 0" ::: "memory");
  #endif
#else
  #pragma unroll
  for (int c = 0; c < CHUNKS; ++c) {
    const int f4 = c * TPB + tid;
    reinterpret_cast<v4f*>(row)[f4] = reinterpret_cast<const v4f*>(gx)[f4];
  }
#endif

  if (tid == 0) {
    int kk = kptr[0];
    if (kk > COLS) kk = COLS;
    if (kk <= 0) { scal[0] = 0xFFFFFFFFu; scal[1] = 0u; scal[2] = 1u; }
    else         { scal[0] = 0u; scal[1] = (unsigned)kk; scal[2] = 0u; }
    scal[3] = 0u;
  }
  __syncthreads();

  const int hid = tid >> 6;  // 1024 threads -> 16 histogram copies

  // ------- 4-round MSB-first radix select over ReLU'd bit patterns -------
  // Positive float bit patterns order identically to their values as uints;
  // everything <= 0 maps to pattern 0 and is excluded from the histograms.
  for (int s = 24; s >= 0; s -= 8) {
    for (int i = tid; i < NHIST * 256; i += TPB) whist[i] = 0u;
    __syncthreads();
    const unsigned done   = scal[2];
    const unsigned prefix = scal[0];
    const unsigned mask   = (s == 24) ? 0u : (0xFFFFFFFFu << (s + 8));
    if (!done) {
      #pragma unroll
      for (int j = 0; j < EPT; ++j) {
        const unsigned u = __float_as_uint(row[j * TPB + tid]);
        const unsigned b = ((int)u > 0) ? u : 0u;
        if (b != 0u && (b & mask) == prefix)
          atomicAdd(&whist[hid * 256 + ((b >> s) & 255u)], 1u);
      }
    }
    __syncthreads();
    if (tid == 0 && !done) {
      const unsigned needL = scal[1];
      unsigned cum = 0;
      int sel = -1;
      for (int b = 255; b >= 0; --b) {
        unsigned cnt = 0;
        for (int h = 0; h < NHIST; ++h) cnt += whist[h * 256 + b];
        if (cum + cnt >= needL) { sel = b; scal[1] = needL - cum; break; }
        cum += cnt;
      }
      if (sel < 0) {  // fewer than k positive values: threshold is zero
        scal[0] = 0u; scal[1] = 0u; scal[2] = 1u;
      } else {
        scal[0] = prefix | ((unsigned)sel << s);
      }
    }
    __syncthreads();
  }

  const unsigned thresh = scal[0];
  const unsigned rem    = scal[1];

  // ---------------- final pass: write kept values, zeros elsewhere ----------------
  #pragma unroll
  for (int c = 0; c < CHUNKS; ++c) {
    const int f4 = c * TPB + tid;
    const v4f v  = reinterpret_cast<const v4f*>(row)[f4];
    v4f ov;
    #pragma unroll
    for (int e = 0; e < 4; ++e) {
      const unsigned u = __float_as_uint(v[e]);
      const unsigned b = ((int)u > 0) ? u : 0u;
      float r = 0.0f;
      if (b > thresh) {
        r = __uint_as_float(b);
      } else if (b == thresh && b != 0u) {
        if (atomicAdd(&scal[3], 1u) < rem) r = __uint_as_float(b);
      }
      ov[e] = r;
    }
    __builtin_nontemporal_store(ov, reinterpret_cast<v4f*>(gout) + f4);
  }
}

extern "C" void kernel_launch(void* const* d_in, const int* in_sizes, int n_in,
                              void* d_out, int out_size, void* d_ws, size_t ws_size,
                              hipStream_t stream) {
  const float* x = (const float*)d_in[0];
  const int*   k = (const int*)d_in[1];
  float*       out = (float*)d_out;
  const int rows = in_sizes[0] / COLS;  // 8192
  const size_t shmem = (size_t)COLS * 4 + NHIST * 256 * 4 + 4 * sizeof(unsigned);
  topk64_relu_kernel<<<rows, TPB, shmem, stream>>>(x, k, out);
  (void)n_in; (void)out_size; (void)d_ws; (void)ws_size;
}